// GAT_20633022889990
// MI455X (gfx1250) — compile-verified
//
#include <hip/hip_runtime.h>
#include <hip/hip_bf16.h>
#include <math.h>

#define NNODES 50000
#define NEDGES 800000
#define NHEADS 4

typedef float v2f __attribute__((ext_vector_type(2)));
typedef float v8f __attribute__((ext_vector_type(8)));

// ---------------------------------------------------------------------------
// utility: fill a u32 buffer with a constant bit pattern (graph-capture safe)
// ---------------------------------------------------------------------------
__global__ void fill_u32_kernel(unsigned int* __restrict__ p, unsigned int v, int n) {
  int i = blockIdx.x * blockDim.x + threadIdx.x;
  if (i < n) p[i] = v;
}

// ---------------------------------------------------------------------------
// BT[c, k] = (c < cols) ? W[k, c] : 0   -- transpose + zero-pad columns so the
// GEMM B-fragment loads are contiguous and unconditional.
// ---------------------------------------------------------------------------
__global__ void pad_transpose_kernel(const float* __restrict__ W, float* __restrict__ BT,
                                     int K, int cols, int colsPad) {
  int i = blockIdx.x * blockDim.x + threadIdx.x;  // c*K + k
  if (i >= colsPad * K) return;
  int c = i / K;
  int k = i - c * K;
  BT[i] = (c < cols) ? W[(size_t)k * cols + c] : 0.f;
}

// ---------------------------------------------------------------------------
// Dense GEMM: C[M=50000, Ncols] = A[M,K] * B[K,Ncols] (B given transposed as
// BT[NcolsPad,K]), fp32, via V_WMMA_F32_16X16X4_F32. One wave per 16x16 output
// tile; block = colTiles waves on one 16-row strip; grid.x = M/16.
//
// ISA 7.12.2 (32-bit A 16x4): lanes 0-15 hold M=lane, {VGPR0,VGPR1}={K+0,K+1};
// lanes 16-31 hold M=lane-16, {K+2,K+3}. B (4x16) mirrored on N.
// C/D: VGPR i -> row (i + 8*(lane>=16)), col = lane&15.
// ---------------------------------------------------------------------------
__global__ __launch_bounds__(384)
void gemm_wmma_f32(const float* __restrict__ A, const float* __restrict__ BT,
                   float* __restrict__ C, int K, int Ncols, int ldc) {
  const int wave  = threadIdx.x >> 5;
  const int lane  = threadIdx.x & 31;
  const int lhalf = lane & 15;          // row (A) / col (B,C) within tile
  const int khalf = (lane >> 4) << 1;   // 0 for lanes 0-15, 2 for lanes 16-31

  const int row0 = blockIdx.x * 16;
  const int col  = wave * 16 + lhalf;   // always < NcolsPad (BT zero-padded)

  const float* Arow = A  + (size_t)(row0 + lhalf) * (size_t)K + khalf;
  const float* Brow = BT + (size_t)col * (size_t)K + khalf;

  v8f acc = {0.f, 0.f, 0.f, 0.f, 0.f, 0.f, 0.f, 0.f};

#pragma unroll 4
  for (int k = 0; k < K; k += 4) {
    v2f a = *reinterpret_cast<const v2f*>(Arow + k);  // global_load_b64
    v2f b = *reinterpret_cast<const v2f*>(Brow + k);  // global_load_b64
    acc = __builtin_amdgcn_wmma_f32_16x16x4_f32(
        /*neg_a=*/false, a, /*neg_b=*/false, b,
        /*c_mod=*/(short)0, acc, /*reuse_a=*/false, /*reuse_b=*/false);
  }

  if (col < Ncols) {
    const int rbase = row0 + ((lane >> 4) << 3);  // +8 for upper half lanes
#pragma unroll
    for (int i = 0; i < 8; i++) {
      C[(size_t)(rbase + i) * (size_t)ldc + col] = acc[i];
    }
  }
}

// ---------------------------------------------------------------------------
// el[n,h] = sum_d feat[n,h,d]*al[h,d];  er likewise. One thread per (n,h).
// ---------------------------------------------------------------------------
__global__ void attn_scores_kernel(const float* __restrict__ feat,
                                   const float* __restrict__ al,
                                   const float* __restrict__ ar,
                                   float* __restrict__ el, float* __restrict__ er,
                                   int D) {
  int idx = blockIdx.x * blockDim.x + threadIdx.x;  // n*H + h
  if (idx >= NNODES * NHEADS) return;
  int h = idx & (NHEADS - 1);
  const float* f  = feat + (size_t)idx * (size_t)D;
  const float* wl = al + h * D;
  const float* wr = ar + h * D;
  float sl = 0.f, sr = 0.f;
  for (int d = 0; d < D; d++) {
    float v = f[d];
    sl += v * wl[d];
    sr += v * wr[d];
  }
  el[idx] = sl;
  er[idx] = sr;
}

// float atomic max via monotonic bit encoding (init sentinel = -inf bits)
__device__ inline void atomicMaxF(float* addr, float v) {
  if (v >= 0.f) atomicMax((int*)addr, __float_as_int(v));
  else          atomicMin((unsigned int*)addr, __float_as_uint(v));
}

// ---------------------------------------------------------------------------
// pass 1: e = leaky_relu(el[src]+er[dst]); segment max over dst
// ---------------------------------------------------------------------------
__global__ void edge_pass1_kernel(const int* __restrict__ src, const int* __restrict__ dst,
                                  const float* __restrict__ el, const float* __restrict__ er,
                                  float* __restrict__ e, float* __restrict__ emax) {
  int i = blockIdx.x * blockDim.x + threadIdx.x;
  if (i >= NEDGES) return;
  int s = src[i], d = dst[i];
  float4 es = *reinterpret_cast<const float4*>(el + (size_t)s * NHEADS);  // b128
  float4 ed = *reinterpret_cast<const float4*>(er + (size_t)d * NHEADS);  // b128
  float4 v;
  v.x = es.x + ed.x; v.y = es.y + ed.y; v.z = es.z + ed.z; v.w = es.w + ed.w;
  v.x = (v.x >= 0.f) ? v.x : 0.2f * v.x;
  v.y = (v.y >= 0.f) ? v.y : 0.2f * v.y;
  v.z = (v.z >= 0.f) ? v.z : 0.2f * v.z;
  v.w = (v.w >= 0.f) ? v.w : 0.2f * v.w;
  *reinterpret_cast<float4*>(e + (size_t)i * NHEADS) = v;
  float* em = emax + (size_t)d * NHEADS;
  atomicMaxF(em + 0, v.x);
  atomicMaxF(em + 1, v.y);
  atomicMaxF(em + 2, v.z);
  atomicMaxF(em + 3, v.w);
}

// ---------------------------------------------------------------------------
// pass 2: e_exp = exp(e - emax[dst]); segment sum -> denom
// ---------------------------------------------------------------------------
__global__ void edge_pass2_kernel(const int* __restrict__ dst,
                                  float* __restrict__ e,
                                  const float* __restrict__ emax,
                                  float* __restrict__ denom) {
  int i = blockIdx.x * blockDim.x + threadIdx.x;
  if (i >= NEDGES) return;
  int d = dst[i];
  float4 v  = *reinterpret_cast<const float4*>(e + (size_t)i * NHEADS);
  float4 em = *reinterpret_cast<const float4*>(emax + (size_t)d * NHEADS);
  v.x = expf(v.x - em.x);
  v.y = expf(v.y - em.y);
  v.z = expf(v.z - em.z);
  v.w = expf(v.w - em.w);
  *reinterpret_cast<float4*>(e + (size_t)i * NHEADS) = v;
  float* dn = denom + (size_t)d * NHEADS;
  atomicAdd(dn + 0, v.x);
  atomicAdd(dn + 1, v.y);
  atomicAdd(dn + 2, v.z);
  atomicAdd(dn + 3, v.w);
}

// ---------------------------------------------------------------------------
// pass 3: agg[dst,h,:] += (e_exp/denom[dst,h]) * feat[src,h,:]
// one thread per (edge, head); float4 gather path when rows are 16B-aligned
// ---------------------------------------------------------------------------
__global__ void edge_pass3_kernel(const int* __restrict__ src, const int* __restrict__ dst,
                                  const float* __restrict__ e, const float* __restrict__ denom,
                                  const float* __restrict__ feat, float* __restrict__ agg,
                                  int D) {
  int i = blockIdx.x * blockDim.x + threadIdx.x;  // edge*H + h
  if (i >= NEDGES * NHEADS) return;
  int eid = i >> 2;
  int h   = i & (NHEADS - 1);
  int s = src[eid], d = dst[eid];
  float alpha = e[i] / denom[d * NHEADS + h];
  const float* fs = feat + ((size_t)s * NHEADS + h) * (size_t)D;
  float*       ad = agg  + ((size_t)d * NHEADS + h) * (size_t)D;
  if ((D & 3) == 0) {
    for (int dd = 0; dd < D; dd += 4) {
      float4 f = *reinterpret_cast<const float4*>(fs + dd);  // b128 gather
      atomicAdd(ad + dd + 0, f.x * alpha);
      atomicAdd(ad + dd + 1, f.y * alpha);
      atomicAdd(ad + dd + 2, f.z * alpha);
      atomicAdd(ad + dd + 3, f.w * alpha);
    }
  } else {
    for (int dd = 0; dd < D; dd++) {
      atomicAdd(ad + dd, fs[dd] * alpha);
    }
  }
}

// ---------------------------------------------------------------------------
// ELU in place (jax.nn.elu: x>0 ? x : expm1(x))
// ---------------------------------------------------------------------------
__global__ void elu_kernel(float* __restrict__ p, int n) {
  int i = blockIdx.x * blockDim.x + threadIdx.x;
  if (i >= n) return;
  float v = p[i];
  p[i] = (v > 0.f) ? v : expm1f(v);
}

// ---------------------------------------------------------------------------
// final: mean over heads (47 classes per head, node stride 188), log_softmax
// ---------------------------------------------------------------------------
__global__ void final_kernel(const float* __restrict__ agg, float* __restrict__ out) {
  int n = blockIdx.x * blockDim.x + threadIdx.x;
  if (n >= NNODES) return;
  const float* a = agg + (size_t)n * 188;
  float v[47];
  float m = -INFINITY;
#pragma unroll 1
  for (int c = 0; c < 47; c++) {
    float t = 0.25f * (a[c] + a[47 + c] + a[94 + c] + a[141 + c]);
    v[c] = t;
    m = fmaxf(m, t);
  }
  float ssum = 0.f;
#pragma unroll 1
  for (int c = 0; c < 47; c++) ssum += expf(v[c] - m);
  float lse = m + logf(ssum);
#pragma unroll 1
  for (int c = 0; c < 47; c++) out[(size_t)n * 47 + c] = v[c] - lse;
}

// ---------------------------------------------------------------------------
// host side
// ---------------------------------------------------------------------------
extern "C" void kernel_launch(void* const* d_in, const int* in_sizes, int n_in,
                              void* d_out, int out_size, void* d_ws, size_t ws_size,
                              hipStream_t stream) {
  const float* x   = (const float*)d_in[0];
  const int*   src = (const int*)  d_in[1];
  const int*   dst = (const int*)  d_in[2];
  const float* W1  = (const float*)d_in[3];
  const float* al1 = (const float*)d_in[4];
  const float* ar1 = (const float*)d_in[5];
  const float* W2  = (const float*)d_in[6];
  const float* al2 = (const float*)d_in[7];
  const float* ar2 = (const float*)d_in[8];
  const float* W3  = (const float*)d_in[9];
  const float* al3 = (const float*)d_in[10];
  const float* ar3 = (const float*)d_in[11];

  char* ws = (char*)d_ws;
  size_t off = 0;
  auto carve = [&](size_t bytes) -> char* {
    char* p = ws + off;
    off = (off + bytes + 255) & ~(size_t)255;
    return p;
  };
  float* feat  = (float*)carve((size_t)NNODES * 192 * sizeof(float));
  float* agg   = (float*)carve((size_t)NNODES * 192 * sizeof(float));
  float* el    = (float*)carve((size_t)NNODES * NHEADS * sizeof(float));
  float* er    = (float*)carve((size_t)NNODES * NHEADS * sizeof(float));
  float* emax  = (float*)carve((size_t)NNODES * NHEADS * sizeof(float));
  float* denom = (float*)carve((size_t)NNODES * NHEADS * sizeof(float));
  float* ebuf  = (float*)carve((size_t)NEDGES * NHEADS * sizeof(float));
  float* BT    = (float*)carve((size_t)192 * 256 * sizeof(float));  // max colsPad*K

  const int rowTiles = NNODES / 16;  // 3125

  auto run_layer = [&](const float* hin, int K, const float* W,
                       const float* al, const float* ar, int D, bool elu_act) {
    const int cols     = NHEADS * D;            // 128 or 188
    const int colsPad  = (cols + 15) & ~15;     // 128 or 192
    const int colTiles = colsPad / 16;          // 8 or 12
    // 0) transpose + zero-pad weights so GEMM loads are unconditional b64s
    {
      int n = colsPad * K;
      pad_transpose_kernel<<<(n + 255) / 256, 256, 0, stream>>>(W, BT, K, cols, colsPad);
    }
    // 1) GEMM: feat = hin @ W
    gemm_wmma_f32<<<rowTiles, colTiles * 32, 0, stream>>>(hin, BT, feat, K, cols, cols);
    // 2) attention scores
    {
      int nth = NNODES * NHEADS;
      attn_scores_kernel<<<(nth + 255) / 256, 256, 0, stream>>>(feat, al, ar, el, er, D);
    }
    // 3) re-init reduction buffers (every call; graph-replay safe)
    {
      int nh = NNODES * NHEADS;
      fill_u32_kernel<<<(nh + 255) / 256, 256, 0, stream>>>((unsigned*)emax, 0xFF800000u, nh);
      fill_u32_kernel<<<(nh + 255) / 256, 256, 0, stream>>>((unsigned*)denom, 0u, nh);
      int na = NNODES * cols;
      fill_u32_kernel<<<(na + 255) / 256, 256, 0, stream>>>((unsigned*)agg, 0u, na);
    }
    // 4) segment softmax + aggregation
    edge_pass1_kernel<<<(NEDGES + 255) / 256, 256, 0, stream>>>(src, dst, el, er, ebuf, emax);
    edge_pass2_kernel<<<(NEDGES + 255) / 256, 256, 0, stream>>>(dst, ebuf, emax, denom);
    {
      int nth = NEDGES * NHEADS;
      edge_pass3_kernel<<<(nth + 255) / 256, 256, 0, stream>>>(src, dst, ebuf, denom, feat, agg, D);
    }
    // 5) activation
    if (elu_act) {
      int na = NNODES * cols;
      elu_kernel<<<(na + 255) / 256, 256, 0, stream>>>(agg, na);
    }
  };

  run_layer(x,   256, W1, al1, ar1, 32, true);   // layer 1
  run_layer(agg, 128, W2, al2, ar2, 32, true);   // layer 2 (agg consumed by GEMM before being zeroed)
  run_layer(agg, 128, W3, al3, ar3, 47, false);  // layer 3

  final_kernel<<<(NNODES + 255) / 256, 256, 0, stream>>>(agg, (float*)d_out);
}